// TritonModel_2078764171678
// MI455X (gfx1250) — compile-verified
//
#include <hip/hip_runtime.h>
#include <hip/hip_bf16.h>

// Streaming causal-conv state-update + SiLU for MI455X (gfx1250).
// Memory-bound (AI ~0.67 FLOP/B): targets 23.3 TB/s HBM streaming.
//  - 128-bit coalesced NT loads for conv_state, NT loads for x, NT stores for y
//  - weight slice (8KB / 1024 channels) staged once per block into LDS with the
//    CDNA5 async global->LDS copy (global_load_async_to_lds_b128 + s_wait_asynccnt),
//    reused across ROWS batch rows per block
//  - 2-D grid (chunk, row-group): no integer division in the kernel prologue
//  - ROWS fully unrolled: 24 global loads in flight per wave before first wait

typedef __attribute__((ext_vector_type(4))) unsigned int u32x4;
typedef __attribute__((ext_vector_type(2))) unsigned int u32x2;

__device__ __forceinline__ float bflo(unsigned int u) {
    return __uint_as_float(u << 16);
}
__device__ __forceinline__ float bfhi(unsigned int u) {
    return __uint_as_float(u & 0xffff0000u);
}
__device__ __forceinline__ unsigned int f2bf_rne(float f) {
    unsigned int u = __float_as_uint(f);
    if ((u & 0x7fffffffu) > 0x7f800000u) return (u >> 16) | 1u;  // NaN stays NaN
    return (u + 0x7fffu + ((u >> 16) & 1u)) >> 16;               // round-to-nearest-even
}
__device__ __forceinline__ float silu(float a) {
    return a * (1.0f / (1.0f + __expf(-a)));
}

#define TPB       256
#define CH_PER_T  4
#define CH_PER_B  (TPB * CH_PER_T)   // 1024 channels per block

template <int ROWS>
__global__ void __launch_bounds__(TPB)
conv_update_silu_kernel(const unsigned short* __restrict__ x,
                        const unsigned short* __restrict__ state,
                        const unsigned short* __restrict__ weight,
                        unsigned short* __restrict__ out,
                        int D)
{
    __shared__ __align__(16) unsigned short wsm[CH_PER_B * 4];  // 8 KB

    const int tid   = threadIdx.x;
    const int chunk = blockIdx.x;          // which 1024-channel slice of D
    const int brow0 = blockIdx.y * ROWS;   // first batch row for this block

    // ---- Stage 8KB weight slice into LDS via CDNA5 async global->LDS copy ----
    // Each of 256 threads copies 2 x 16B; the second copy uses offset:4096,
    // which is applied to both the global and the LDS address.
    {
        unsigned int       lds  = ((unsigned int)(unsigned long long)(&wsm[0])) + (unsigned)tid * 16u;
        unsigned long long gsrc = (unsigned long long)(weight + (size_t)chunk * CH_PER_B * 4)
                                  + (unsigned long long)tid * 16ull;
        asm volatile(
            "global_load_async_to_lds_b128 %0, %1, off\n\t"
            "global_load_async_to_lds_b128 %0, %1, off offset:4096\n\t"
            :: "v"(lds), "v"(gsrc) : "memory");
        asm volatile("s_wait_asynccnt 0x0" ::: "memory");
    }
    __syncthreads();

    // Per-thread weights: 4 channels * 4 bf16 = 32B from LDS (2x ds_load_b128).
    const u32x4* wp = (const u32x4*)(&wsm[tid * 16]);
    const u32x4  wA = wp[0];  // channels c0,c1 : dwords pack (w0,w1)(w2,w3)
    const u32x4  wB = wp[1];  // channels c2,c3

    const int d0 = chunk * CH_PER_B + tid * CH_PER_T;

#pragma unroll
    for (int r = 0; r < ROWS; ++r) {
        const size_t row = (size_t)(brow0 + r) * (size_t)D;

        // 32B of conv_state (two b128 NT loads), 8B of x (b64 NT load)
        const u32x4* sp  = (const u32x4*)(state + (row + (size_t)d0) * 4);
        u32x4 s01 = __builtin_nontemporal_load(sp);
        u32x4 s23 = __builtin_nontemporal_load(sp + 1);
        u32x2 xv  = __builtin_nontemporal_load((const u32x2*)(x + row + (size_t)d0));

        float xs0 = bflo(xv.x), xs1 = bfhi(xv.x), xs2 = bflo(xv.y), xs3 = bfhi(xv.y);

        // new_state = [s1, s2, s3, x];  acc = w0*s1 + w1*s2 + w2*s3 + w3*x  (fp32)
        float a0 = bflo(wA.x)*bfhi(s01.x) + bfhi(wA.x)*bflo(s01.y) + bflo(wA.y)*bfhi(s01.y) + bfhi(wA.y)*xs0;
        float a1 = bflo(wA.z)*bfhi(s01.z) + bfhi(wA.z)*bflo(s01.w) + bflo(wA.w)*bfhi(s01.w) + bfhi(wA.w)*xs1;
        float a2 = bflo(wB.x)*bfhi(s23.x) + bfhi(wB.x)*bflo(s23.y) + bflo(wB.y)*bfhi(s23.y) + bfhi(wB.y)*xs2;
        float a3 = bflo(wB.z)*bfhi(s23.z) + bfhi(wB.z)*bflo(s23.w) + bflo(wB.w)*bfhi(s23.w) + bfhi(wB.w)*xs3;

        u32x2 o;
        o.x = f2bf_rne(silu(a0)) | (f2bf_rne(silu(a1)) << 16);
        o.y = f2bf_rne(silu(a2)) | (f2bf_rne(silu(a3)) << 16);
        __builtin_nontemporal_store(o, (u32x2*)(out + row + (size_t)d0));
    }
}

extern "C" void kernel_launch(void* const* d_in, const int* in_sizes, int n_in,
                              void* d_out, int out_size, void* d_ws, size_t ws_size,
                              hipStream_t stream) {
    (void)n_in; (void)out_size; (void)d_ws; (void)ws_size;
    const unsigned short* x      = (const unsigned short*)d_in[0];  // [B, D] bf16
    const unsigned short* state  = (const unsigned short*)d_in[1];  // [B, D, 4] bf16
    const unsigned short* weight = (const unsigned short*)d_in[2];  // [D, 4] bf16
    unsigned short*       out    = (unsigned short*)d_out;          // [B, D] bf16

    const int K  = 4;
    const int D  = in_sizes[2] / K;              // 10240
    const long long n = (long long)in_sizes[0];  // B*D
    const int B  = (int)(n / D);                 // 4096
    const int nchunk = D / CH_PER_B;             // 10 (D is a multiple of 1024)

    if ((B & 7) == 0) {
        dim3 grid(nchunk, B / 8);                // 10 x 512 = 5120 blocks
        conv_update_silu_kernel<8><<<grid, TPB, 0, stream>>>(x, state, weight, out, D);
    } else if ((B & 3) == 0) {
        dim3 grid(nchunk, B / 4);
        conv_update_silu_kernel<4><<<grid, TPB, 0, stream>>>(x, state, weight, out, D);
    } else if ((B & 1) == 0) {
        dim3 grid(nchunk, B / 2);
        conv_update_silu_kernel<2><<<grid, TPB, 0, stream>>>(x, state, weight, out, D);
    } else {
        dim3 grid(nchunk, B);
        conv_update_silu_kernel<1><<<grid, TPB, 0, stream>>>(x, state, weight, out, D);
    }
}